// KvCache_16621523436389
// MI455X (gfx1250) — compile-verified
//
#include <hip/hip_runtime.h>
#include <stdint.h>

// KV-cache append for MI455X (gfx1250).
//
// Pure data movement: 256 MiB read + 256 MiB write => ~22 us at 23.3 TB/s.
// No FLOPs => no WMMA; the CDNA5 feature that fits is the async global<->LDS
// DMA path (ASYNCcnt-tracked): payload never touches VGPRs.
//
// Each block copies ROWS=4 consecutive (b,p) rows (4 KiB each, K and V):
//   8x global_load_async_to_lds_b128   (8 DMAs in flight per wave)
//   s_wait_asynccnt 0                  (wave-local; LDS slots lane-private)
//   8x global_store_async_from_lds_b128
// Rows inside [len[b], len[b]+new_len[b]) source from new_keys/new_values
// (the reference's specialized nl<=1 branch is subsumed by the general path).
// All control state (b, p, l, nl, bases) is block-uniform => pure SALU.

namespace {
constexpr int  kB   = 8;
constexpr int  kL   = 4096;
constexpr int  kH   = 8;
constexpr int  kD   = 128;
constexpr int  kT   = 128;
constexpr int  kRow = kH * kD;                       // 1024 floats / row
constexpr long kKV  = (long)kB * kL * kRow;          // 33,554,432 floats
constexpr int  kRows = 4;                            // rows per block
}  // namespace

// Builtin signature (probe-confirmed): (int4 AS1*, int4 AS3*, Ii, Ii).
typedef int int4v __attribute__((ext_vector_type(4)));
typedef __attribute__((address_space(1))) int4v gint4;   // global int4
typedef __attribute__((address_space(3))) int4v lint4;   // LDS int4

__global__ __launch_bounds__(256) void kv_cache_update_async(
    const float* __restrict__ keys, const float* __restrict__ vals,
    const int*   __restrict__ lens,
    const float* __restrict__ nkeys, const float* __restrict__ nvals,
    const int*   __restrict__ nlens,
    float* __restrict__ okeys, float* __restrict__ ovals,
    int*   __restrict__ olens)
{
    // Staging: 4 rows x (4 KiB K + 4 KiB V) = 32 KiB => 10 blocks/WGP.
    __shared__ alignas(16) float ldsK[kRows][256 * 4];
    __shared__ alignas(16) float ldsV[kRows][256 * 4];

    const int row0 = blockIdx.x * kRows;   // 4096 % 4 == 0 -> same b for all
    const int b    = row0 >> 12;           // L == 4096
    const int t    = threadIdx.x;          // 0..255 -> one float4 per row

    // Block-uniform -> scalar loads.
    const int l  = lens[b];
    const int nl = nlens[b];

    // Classify each row's source (uniform scalar selects).
    const float* srcK[kRows];
    const float* srcV[kRows];
#pragma unroll
    for (int r = 0; r < kRows; ++r) {
        const int p = (row0 + r) & (kL - 1);
        if (p >= l && p < l + nl) {
            // Overlay region: pull from the new tokens (p - l < nl <= T).
            const long sb = ((long)b * kT + (p - l)) * kRow;
            srcK[r] = nkeys + sb;
            srcV[r] = nvals + sb;
        } else {
            const long sb = (long)(row0 + r) * kRow;
            srcK[r] = keys + sb;
            srcV[r] = vals + sb;
        }
    }

    // Phase 1: 8 async DMAs global -> LDS (per-lane 16B each).
#pragma unroll
    for (int r = 0; r < kRows; ++r) {
        __builtin_amdgcn_global_load_async_to_lds_b128(
            (gint4*)(srcK[r] + 4 * t), (lint4*)(&ldsK[r][4 * t]), 0, 0);
        __builtin_amdgcn_global_load_async_to_lds_b128(
            (gint4*)(srcV[r] + 4 * t), (lint4*)(&ldsV[r][4 * t]), 0, 0);
    }

    // LDS slots are lane-private, so a wave-local asynccnt wait suffices
    // (no workgroup barrier needed) before the stores read LDS back.
    __builtin_amdgcn_s_wait_asynccnt(0);

    // Phase 2: 8 async DMAs LDS -> global. S_ENDPGM's implicit wait-idle
    // drains these before the kernel is reported complete.
#pragma unroll
    for (int r = 0; r < kRows; ++r) {
        const long ob = (long)(row0 + r) * kRow;
        __builtin_amdgcn_global_store_async_from_lds_b128(
            (gint4*)(okeys + ob + 4 * t), (lint4*)(&ldsK[r][4 * t]), 0, 0);
        __builtin_amdgcn_global_store_async_from_lds_b128(
            (gint4*)(ovals + ob + 4 * t), (lint4*)(&ldsV[r][4 * t]), 0, 0);
    }

    // Third output: updated_lengths = lengths + new_lengths (int32 bits
    // written into the tail of d_out).
    if (row0 == 0 && t < kB) {
        olens[t] = lens[t] + nlens[t];
    }
}

extern "C" void kernel_launch(void* const* d_in, const int* in_sizes, int n_in,
                              void* d_out, int out_size, void* d_ws, size_t ws_size,
                              hipStream_t stream) {
    (void)in_sizes; (void)n_in; (void)out_size; (void)d_ws; (void)ws_size;

    const float* keys  = (const float*)d_in[0];
    const float* vals  = (const float*)d_in[1];
    const int*   lens  = (const int*)d_in[2];
    const float* nkeys = (const float*)d_in[3];
    const float* nvals = (const float*)d_in[4];
    const int*   nlens = (const int*)d_in[5];

    float* okeys = (float*)d_out;
    float* ovals = okeys + kKV;
    int*   olens = (int*)(ovals + kKV);

    kv_cache_update_async<<<(kB * kL) / kRows, 256, 0, stream>>>(
        keys, vals, lens, nkeys, nvals, nlens, okeys, ovals, olens);
}